// ROIAlign_17721035063738
// MI455X (gfx1250) — compile-verified
//
#include <hip/hip_runtime.h>

#define THREADS 64   // 2 waves (wave32); 64 lanes x float4 = 256 channels
#define DEPTH   4    // async pipeline depth (buffers); prefetch distance 3

__global__ __launch_bounds__(THREADS, 1) void roi_align_async_kernel(
    const float* __restrict__ inputs,      // [8,200,200,256]
    const float* __restrict__ boxes,       // [N,4] cy,cx,h,w
    const int*   __restrict__ assignments, // [N]
    float* __restrict__ out)               // [N,7,7,256]
{
    constexpr int FH = 200, FW = 200;
    constexpr int PIX_BYTES = 256 * 4;     // one pixel's channel vector

    // DEPTH buffers x 4 corners x 64 lanes x 16B = 16KB LDS (of 320KB/WGP)
    __shared__ float4 lbuf[DEPTH][4][THREADS];

    const int n   = blockIdx.x;
    const int tid = threadIdx.x;

    const float cy = boxes[n * 4 + 0];
    const float cx = boxes[n * 4 + 1];
    const float bh = boxes[n * 4 + 2];
    const float bw = boxes[n * 4 + 3];
    const int   bi = assignments[n];

    const float y1 = cy - 0.5f * bh, y2 = cy + 0.5f * bh;
    const float x1 = cx - 0.5f * bw, x2 = cx + 0.5f * bw;
    const float binh = bh * (1.0f / 7.0f);
    const float binw = bw * (1.0f / 7.0f);
    const float gy1 = (y1 + 0.25f * binh) * (1.0f / 200.0f);
    const float gy2 = (y2 - 0.25f * binh) * (1.0f / 200.0f);
    const float gx1 = (x1 + 0.25f * binw) * (1.0f / 200.0f);
    const float gx2 = (x2 - 0.25f * binw) * (1.0f / 200.0f);
    const float dgy = gy2 - gy1;
    const float dgx = gx2 - gx1;

    // per-lane base: image base + this lane's 16B channel slice
    const unsigned long long inbase =
        (unsigned long long)inputs + (unsigned long long)(unsigned)(tid * 16);
    const unsigned rowbase = (unsigned)(bi * FH);

    // Issue 4 async corner gathers (global -> LDS, 128b per lane) for sample
    // (i2, j2) into pipeline buffer p; records bilinear weights + valid mask.
    auto issue = [&](int i2, int j2, int p, float& wy, float& wx, float& vf) {
        float ysv = (gy1 + dgy * ((float)i2 * (1.0f / 13.0f))) * 199.0f;
        float xsv = (gx1 + dgx * ((float)j2 * (1.0f / 13.0f))) * 199.0f;
        vf = (ysv >= 0.0f && ysv <= 199.0f && xsv >= 0.0f && xsv <= 199.0f)
                 ? 1.0f : 0.0f;
        float y0f = floorf(ysv);
        float x0f = floorf(xsv);
        wy = ysv - y0f;
        wx = xsv - x0f;
        int y0  = (int)y0f;      y0  = y0  < 0 ? 0 : (y0  > FH - 1 ? FH - 1 : y0);
        int y1c = (int)y0f + 1;  y1c = y1c < 0 ? 0 : (y1c > FH - 1 ? FH - 1 : y1c);
        int x0  = (int)x0f;      x0  = x0  < 0 ? 0 : (x0  > FW - 1 ? FW - 1 : x0);
        int x1c = (int)x0f + 1;  x1c = x1c < 0 ? 0 : (x1c > FW - 1 ? FW - 1 : x1c);
        unsigned r0 = (rowbase + (unsigned)y0)  * (unsigned)FW;
        unsigned r1 = (rowbase + (unsigned)y1c) * (unsigned)FW;
        unsigned long long a00 = inbase + (unsigned long long)((r0 + (unsigned)x0)  * (unsigned)PIX_BYTES);
        unsigned long long a01 = inbase + (unsigned long long)((r0 + (unsigned)x1c) * (unsigned)PIX_BYTES);
        unsigned long long a10 = inbase + (unsigned long long)((r1 + (unsigned)x0)  * (unsigned)PIX_BYTES);
        unsigned long long a11 = inbase + (unsigned long long)((r1 + (unsigned)x1c) * (unsigned)PIX_BYTES);
        unsigned l00 = (unsigned)(size_t)&lbuf[p][0][tid];
        unsigned l01 = (unsigned)(size_t)&lbuf[p][1][tid];
        unsigned l10 = (unsigned)(size_t)&lbuf[p][2][tid];
        unsigned l11 = (unsigned)(size_t)&lbuf[p][3][tid];
        // drain prior ds_reads of this (reused) buffer before the async engine
        // may overwrite it (DS vs async ordering is not guaranteed by HW)
        asm volatile("s_wait_dscnt 0x0" ::: "memory");
        asm volatile("global_load_async_to_lds_b128 %0, %1, off" :: "v"(l00), "v"(a00) : "memory");
        asm volatile("global_load_async_to_lds_b128 %0, %1, off" :: "v"(l01), "v"(a01) : "memory");
        asm volatile("global_load_async_to_lds_b128 %0, %1, off" :: "v"(l10), "v"(a10) : "memory");
        asm volatile("global_load_async_to_lds_b128 %0, %1, off" :: "v"(l11), "v"(a11) : "memory");
    };

    // Rotating weight queue for in-flight samples (wave-uniform values).
    // Invariant: sample s lives in buffer/queue slot (s & 3), and since each
    // oy-row is 28 samples (28 % 4 == 0), slot == (dy*14 + j) & 3 at compile
    // time once the 28-sample body is unrolled.
    float wyq[DEPTH], wxq[DEPTH], vfq[DEPTH];

    // Prologue: prefetch samples 0,1,2 into buffers 0,1,2.
    issue(0, 0, 0, wyq[0], wxq[0], vfq[0]);
    issue(0, 1, 1, wyq[1], wxq[1], vfq[1]);
    issue(0, 2, 2, wyq[2], wxq[2], vfq[2]);

    float4* out4 = (float4*)out + (size_t)n * 49 * 64 + tid;

#pragma unroll 1           // keep code size bounded; 28-sample body unrolls below
    for (int oy = 0; oy < 7; ++oy) {
        float4 acc[7];
#pragma unroll
        for (int k = 0; k < 7; ++k) acc[k] = make_float4(0.f, 0.f, 0.f, 0.f);

#pragma unroll
        for (int dy = 0; dy < 2; ++dy) {
#pragma unroll
            for (int j = 0; j < 14; ++j) {
                const int r  = dy * 14 + j;     // 0..27, compile-time
                const int qs = r & 3;           // slot of current sample
                const int qn = (r + 3) & 3;     // slot being prefetched
                const int di = (r + 3) / 14;    // 0,1,2 (compile-time)
                const int j3 = (r + 3) % 14;    // compile-time
                // Prefetch sample s+3. Past the grid end (only oy==6, r>=25)
                // this clamps to row i=13: an in-bounds dummy whose buffer
                // slot is provably never read again -> no tail branches.
                int i3 = 2 * oy + di;
                if (i3 > 13) i3 = 13;
                issue(i3, j3, qn, wyq[qn], wxq[qn], vfq[qn]);
                // 4 groups in flight after issue; <=12 outstanding (in-order
                // completion) => current sample's 4 gathers are in LDS
                asm volatile("s_wait_asynccnt 0xc" ::: "memory");

                float4 v00 = lbuf[qs][0][tid];
                float4 v01 = lbuf[qs][1][tid];
                float4 v10 = lbuf[qs][2][tid];
                float4 v11 = lbuf[qs][3][tid];
                const float wyc = wyq[qs], wxc = wxq[qs], vfc = vfq[qs];
                float tx, bx;
                float4 sm;
                tx = v00.x + (v01.x - v00.x) * wxc;  bx = v10.x + (v11.x - v10.x) * wxc;
                sm.x = (tx + (bx - tx) * wyc) * vfc;
                tx = v00.y + (v01.y - v00.y) * wxc;  bx = v10.y + (v11.y - v10.y) * wxc;
                sm.y = (tx + (bx - tx) * wyc) * vfc;
                tx = v00.z + (v01.z - v00.z) * wxc;  bx = v10.z + (v11.z - v10.z) * wxc;
                sm.z = (tx + (bx - tx) * wyc) * vfc;
                tx = v00.w + (v01.w - v00.w) * wxc;  bx = v10.w + (v11.w - v10.w) * wxc;
                sm.w = (tx + (bx - tx) * wyc) * vfc;
                const int ox = j >> 1;          // compile-time -> acc in VGPRs
                acc[ox].x += sm.x; acc[ox].y += sm.y;
                acc[ox].z += sm.z; acc[ox].w += sm.w;
            }
        }
#pragma unroll
        for (int k = 0; k < 7; ++k) {
            float4 w = acc[k];
            w.x *= 0.25f; w.y *= 0.25f; w.z *= 0.25f; w.w *= 0.25f;
            out4[(oy * 7 + k) * 64] = w;        // 2x2 mean pool
        }
    }
    // Trailing dummy prefetches are drained by S_ENDPGM's implicit wait-idle.
}

extern "C" void kernel_launch(void* const* d_in, const int* in_sizes, int n_in,
                              void* d_out, int out_size, void* d_ws, size_t ws_size,
                              hipStream_t stream) {
    const float* inputs      = (const float*)d_in[0];
    const float* boxes       = (const float*)d_in[1];
    const int*   assignments = (const int*)d_in[2];
    float* out = (float*)d_out;
    const int N = in_sizes[2];   // one assignment per box
    roi_align_async_kernel<<<dim3(N), dim3(THREADS), 0, stream>>>(
        inputs, boxes, assignments, out);
}